// RNN_45526653338120
// MI455X (gfx1250) — compile-verified
//
#include <hip/hip_runtime.h>
#include <stdint.h>

#define TT 512
#define BB 16384
#define CT 16                    // timesteps staged per chunk
#define NCHUNK (TT / CT)         // 32
#define ROWF (CT * 6)            // 96 floats of x per batch-row per chunk
#define SEGS (ROWF / 4)          // 24 16B segments per row
#define ROWSTRIDE 100            // padded row stride in floats (25*16B, bank-friendly)
#define BUFF (16 * ROWSTRIDE)    // 1600 floats per buffer (16 batches per wave)
#define OPS 12                   // async b128 instructions per lane per chunk (16*24/32)

// ---- CDNA5 async copy: global -> LDS, tracked by ASYNCcnt ----
__device__ __forceinline__ void async_b128(uint32_t lds, const float* g) {
    asm volatile("global_load_async_to_lds_b128 %0, %1, off"
                 :: "v"(lds), "v"(g) : "memory");
}

__device__ __forceinline__ float sig(float x) {
    // 1/(1+exp(-x)) via v_exp_f32 + v_rcp_f32
    return __builtin_amdgcn_rcpf(1.0f + __builtin_amdgcn_exp2f(x * -1.442695040888963f));
}
__device__ __forceinline__ float tnh(float x) {
    // tanh(x) = 2/(1+exp(-2x)) - 1
    return 2.0f * __builtin_amdgcn_rcpf(1.0f + __builtin_amdgcn_exp2f(x * -2.885390081777927f)) - 1.0f;
}

__global__ __launch_bounds__(32) void gru2_fused_kernel(
    const float* __restrict__ encoded, const float* __restrict__ conc,
    const float* __restrict__ w_ih0, const float* __restrict__ w_hh0,
    const float* __restrict__ b_ih0, const float* __restrict__ b_hh0,
    const float* __restrict__ w_ih1, const float* __restrict__ w_hh1,
    const float* __restrict__ b_ih1, const float* __restrict__ b_hh1,
    const float* __restrict__ bn_gamma, const float* __restrict__ bn_beta,
    const float* __restrict__ bn_mean, const float* __restrict__ bn_var,
    const float* __restrict__ fc_w, const float* __restrict__ fc_b,
    float* __restrict__ out)
{
    __shared__ __align__(16) float smem[2 * BUFF];   // 12.8 KB, one wave per block

    const int lane   = threadIdx.x;       // 0..31
    const int p      = lane & 1;          // lane-pair half: owns hidden units {2p, 2p+1}
    const int bl     = lane >> 1;         // batch-in-wave 0..15
    const int batch0 = blockIdx.x * 16;   // first batch of this wave
    const int b      = batch0 + bl;
    const uint32_t smemBase = (uint32_t)(uintptr_t)&smem[0];

    // h-vector lane order: [own0, own1, partner0, partner1] = hidden idx [2p,2p+1,2-2p,3-2p]
    int idx[4];
    idx[0] = 2 * p; idx[1] = 2 * p + 1; idx[2] = 2 - 2 * p; idx[3] = 3 - 2 * p;

    // Per-lane weights (vary with p -> live in VGPRs), hh/ih1 pre-permuted by idx[].
    float wir0[2][6], wiz0[2][6], win0[2][6];
    float whr0[2][4], whz0[2][4], whn0[2][4];
    float wir1[2][4], wiz1[2][4], win1[2][4];
    float whr1[2][4], whz1[2][4], whn1[2][4];
    float br0[2], bz0[2], bni0[2], bnh0[2];
    float br1[2], bz1[2], bni1[2], bnh1[2];

    #pragma unroll
    for (int u = 0; u < 2; ++u) {
        const int hu = 2 * p + u;
        #pragma unroll
        for (int i = 0; i < 6; ++i) {
            wir0[u][i] = w_ih0[(0 + hu) * 6 + i];
            wiz0[u][i] = w_ih0[(4 + hu) * 6 + i];
            win0[u][i] = w_ih0[(8 + hu) * 6 + i];
        }
        #pragma unroll
        for (int j = 0; j < 4; ++j) {
            whr0[u][j] = w_hh0[(0 + hu) * 4 + idx[j]];
            whz0[u][j] = w_hh0[(4 + hu) * 4 + idx[j]];
            whn0[u][j] = w_hh0[(8 + hu) * 4 + idx[j]];
            wir1[u][j] = w_ih1[(0 + hu) * 4 + idx[j]];
            wiz1[u][j] = w_ih1[(4 + hu) * 4 + idx[j]];
            win1[u][j] = w_ih1[(8 + hu) * 4 + idx[j]];
            whr1[u][j] = w_hh1[(0 + hu) * 4 + idx[j]];
            whz1[u][j] = w_hh1[(4 + hu) * 4 + idx[j]];
            whn1[u][j] = w_hh1[(8 + hu) * 4 + idx[j]];
        }
        br0[u]  = b_ih0[hu]     + b_hh0[hu];
        bz0[u]  = b_ih0[4 + hu] + b_hh0[4 + hu];
        bni0[u] = b_ih0[8 + hu];  bnh0[u] = b_hh0[8 + hu];
        br1[u]  = b_ih1[hu]     + b_hh1[hu];
        bz1[u]  = b_ih1[4 + hu] + b_hh1[4 + hu];
        bni1[u] = b_ih1[8 + hu];  bnh1[u] = b_hh1[8 + hu];
    }

    float h0v[4] = {0.f, 0.f, 0.f, 0.f};
    float h1v[4] = {0.f, 0.f, 0.f, 0.f};

    // Stage one chunk (16 steps x 16 batches) into LDS buffer (c&1) via async copies.
    auto issue = [&](int c) {
        #pragma unroll
        for (int m = 0; m < OPS; ++m) {
            const int flat = m * 32 + lane;          // 0..383
            const int row  = flat / SEGS;            // batch-in-wave
            const int seg  = flat - row * SEGS;      // 16B segment within row-chunk
            const float* g = encoded + (size_t)(batch0 + row) * (TT * 6)
                                     + (size_t)c * ROWF + seg * 4;
            const uint32_t l = smemBase
                + (uint32_t)(((c & 1) * BUFF + row * ROWSTRIDE + seg * 4) * 4);
            async_b128(l, g);
        }
    };

    issue(0);
    issue(1);

    for (int c = 0; c < NCHUNK; ++c) {
        // Async completions are in-order: waiting <=12 guarantees chunk c landed.
        if (c == NCHUNK - 1) asm volatile("s_wait_asynccnt 0"  ::: "memory");
        else                 asm volatile("s_wait_asynccnt 12" ::: "memory");

        const float* xr = &smem[(c & 1) * BUFF + bl * ROWSTRIDE];
        for (int t = 0; t < CT; ++t) {
            float x[6];
            #pragma unroll
            for (int i = 0; i < 6; ++i) x[i] = xr[t * 6 + i];  // pair-broadcast DS reads

            float nh[2];
            // ---- layer 0 ----
            #pragma unroll
            for (int u = 0; u < 2; ++u) {
                float gr = br0[u], gz = bz0[u], gi = bni0[u], gh = bnh0[u];
                #pragma unroll
                for (int i = 0; i < 6; ++i) {
                    gr = fmaf(wir0[u][i], x[i], gr);
                    gz = fmaf(wiz0[u][i], x[i], gz);
                    gi = fmaf(win0[u][i], x[i], gi);
                }
                #pragma unroll
                for (int j = 0; j < 4; ++j) {
                    gr = fmaf(whr0[u][j], h0v[j], gr);
                    gz = fmaf(whz0[u][j], h0v[j], gz);
                    gh = fmaf(whn0[u][j], h0v[j], gh);
                }
                const float r = sig(gr), z = sig(gz);
                const float n = tnh(fmaf(r, gh, gi));
                nh[u] = n + z * (h0v[u] - n);       // (1-z)*n + z*h
            }
            h0v[0] = nh[0]; h0v[1] = nh[1];
            h0v[2] = __shfl_xor(nh[0], 1, 32);
            h0v[3] = __shfl_xor(nh[1], 1, 32);

            // ---- layer 1 (input = new h0, already in idx order) ----
            #pragma unroll
            for (int u = 0; u < 2; ++u) {
                float gr = br1[u], gz = bz1[u], gi = bni1[u], gh = bnh1[u];
                #pragma unroll
                for (int j = 0; j < 4; ++j) {
                    gr = fmaf(wir1[u][j], h0v[j], gr);
                    gz = fmaf(wiz1[u][j], h0v[j], gz);
                    gi = fmaf(win1[u][j], h0v[j], gi);
                }
                #pragma unroll
                for (int j = 0; j < 4; ++j) {
                    gr = fmaf(whr1[u][j], h1v[j], gr);
                    gz = fmaf(whz1[u][j], h1v[j], gz);
                    gh = fmaf(whn1[u][j], h1v[j], gh);
                }
                const float r = sig(gr), z = sig(gz);
                const float n = tnh(fmaf(r, gh, gi));
                nh[u] = n + z * (h1v[u] - n);
            }
            h1v[0] = nh[0]; h1v[1] = nh[1];
            h1v[2] = __shfl_xor(nh[0], 1, 32);
            h1v[3] = __shfl_xor(nh[1], 1, 32);
        }
        if (c + 2 < NCHUNK) issue(c + 2);          // prefetch next-next chunk
    }

    // ---- LeakyReLU -> BatchNorm(eval) -> FC (with conc concat) ----
    float part = 0.0f;
    #pragma unroll
    for (int u = 0; u < 2; ++u) {
        const int k = 2 * p + u;
        float h = h1v[u];
        h = (h >= 0.0f) ? h : 0.2f * h;
        h = (h - bn_mean[k]) * __builtin_amdgcn_rsqf(bn_var[k] + 1e-5f) * bn_gamma[k]
            + bn_beta[k];
        part = fmaf(fc_w[k], h, part);
    }
    part += __shfl_xor(part, 1, 32);
    if (p == 0) {
        out[b] = fmaf(fc_w[4], conc[b], part + fc_b[0]);
    }
}

extern "C" void kernel_launch(void* const* d_in, const int* in_sizes, int n_in,
                              void* d_out, int out_size, void* d_ws, size_t ws_size,
                              hipStream_t stream) {
    (void)in_sizes; (void)n_in; (void)out_size; (void)d_ws; (void)ws_size;
    const float* encoded  = (const float*)d_in[0];
    const float* conc     = (const float*)d_in[1];
    const float* w_ih0    = (const float*)d_in[2];
    const float* w_hh0    = (const float*)d_in[3];
    const float* b_ih0    = (const float*)d_in[4];
    const float* b_hh0    = (const float*)d_in[5];
    const float* w_ih1    = (const float*)d_in[6];
    const float* w_hh1    = (const float*)d_in[7];
    const float* b_ih1    = (const float*)d_in[8];
    const float* b_hh1    = (const float*)d_in[9];
    const float* bn_gamma = (const float*)d_in[10];
    const float* bn_beta  = (const float*)d_in[11];
    const float* bn_mean  = (const float*)d_in[12];
    const float* bn_var   = (const float*)d_in[13];
    const float* fc_w     = (const float*)d_in[14];
    const float* fc_b     = (const float*)d_in[15];
    float* out = (float*)d_out;

    dim3 grid(BB / 16);   // 1024 single-wave workgroups: 16 batches per wave, 2 lanes/batch
    dim3 block(32);
    gru2_fused_kernel<<<grid, block, 0, stream>>>(
        encoded, conc, w_ih0, w_hh0, b_ih0, b_hh0,
        w_ih1, w_hh1, b_ih1, b_hh1,
        bn_gamma, bn_beta, bn_mean, bn_var, fc_w, fc_b, out);
}